// GCN_5987184410903
// MI455X (gfx1250) — compile-verified
//
#include <hip/hip_runtime.h>
#include <math.h>

typedef __attribute__((ext_vector_type(2))) float v2f;
typedef __attribute__((ext_vector_type(8))) float v8f;

// ---------------------------------------------------------------------------
// Small utility kernels
// ---------------------------------------------------------------------------
__global__ void fill_f32(float* __restrict__ p, float v, int n) {
    int i = blockIdx.x * blockDim.x + threadIdx.x;
    if (i < n) p[i] = v;
}

__global__ void deg_accum(const int* __restrict__ dst, float* __restrict__ deg, int E) {
    int i = blockIdx.x * blockDim.x + threadIdx.x;
    if (i < E) atomicAdd(&deg[dst[i]], 1.0f);
}

__global__ void deg_rsqrt(float* __restrict__ deg, int n) {
    int i = blockIdx.x * blockDim.x + threadIdx.x;
    if (i < n) {
        float d = deg[i];
        deg[i] = d > 0.0f ? rsqrtf(d) : 0.0f;
    }
}

// ---------------------------------------------------------------------------
// WMMA f32 GEMM:  out[nrows x (NT*16)] = A[nrows x K] * W[K x (NT*16)] (+bias)
// One wave computes a 16-row tile with NT 16x16 accumulators, stepping K by 4
// using V_WMMA_F32_16X16X4_F32.
//
// A-fragment layout (16x4 f32, ISA 7.12.2): lanes 0-15 VGPR0=K0,VGPR1=K1 ;
// lanes 16-31 VGPR0=K2,VGPR1=K3 (same M=lane&15). B mirrored (N=lane&15).
// C/D: VGPR r -> M = r + 8*(lane>=16), N = lane&15.
// ---------------------------------------------------------------------------
template <int K, int NT>
__global__ void gemm_wmma_f32(const float* __restrict__ A,
                              const float* __restrict__ W,
                              const float* __restrict__ bias,
                              float* __restrict__ out,
                              int nrows) {
    constexpr int NCOLS = NT * 16;
    const int lane = threadIdx.x & 31;
    const int wave = threadIdx.x >> 5;
    const int tile = blockIdx.x * (blockDim.x >> 5) + wave;
    if (tile * 16 >= nrows) return;

    const int half = lane >> 4;   // 0: K pair {0,1}, 1: K pair {2,3}
    const int m    = lane & 15;   // row within tile / col within tile

    int arow = tile * 16 + m;
    if (arow >= nrows) arow = nrows - 1;   // clamp (junk rows never stored)
    const float* __restrict__ Arow = A + (size_t)arow * K;

    v8f acc[NT];
#pragma unroll
    for (int t = 0; t < NT; ++t) acc[t] = (v8f){0.f, 0.f, 0.f, 0.f, 0.f, 0.f, 0.f, 0.f};

#pragma unroll
    for (int k0 = 0; k0 < K; k0 += 4) {
        const int ka = k0 + 2 * half;
        v2f a;
        a.x = Arow[ka];
        a.y = Arow[ka + 1];
#pragma unroll
        for (int t = 0; t < NT; ++t) {
            const int n = t * 16 + m;
            v2f b;
            b.x = W[(size_t)ka * NCOLS + n];
            b.y = W[(size_t)(ka + 1) * NCOLS + n];
            acc[t] = __builtin_amdgcn_wmma_f32_16x16x4_f32(
                /*neg_a=*/false, a, /*neg_b=*/false, b,
                /*c_mod=*/(short)0, acc[t], /*reuse_a=*/false, /*reuse_b=*/false);
        }
    }

#pragma unroll
    for (int t = 0; t < NT; ++t) {
#pragma unroll
        for (int r = 0; r < 8; ++r) {
            const int row = tile * 16 + r + 8 * half;
            const int col = t * 16 + m;
            if (row < nrows) {
                float v = acc[t][r];
                if (bias) v += bias[col];
                out[(size_t)row * NCOLS + col] = v;
            }
        }
    }
}

// ---------------------------------------------------------------------------
// GCN aggregation, H == 32: one wave per edge, lane = feature.
// out[dst] += xw[src] * dis[src]*dis[dst]; edges e >= E are self-loops.
// ---------------------------------------------------------------------------
__global__ void aggregate32(const float* __restrict__ xw,
                            const int* __restrict__ src,
                            const int* __restrict__ dst,
                            const float* __restrict__ dis,
                            float* __restrict__ out,
                            int E, int N) {
    const int e = blockIdx.x * (blockDim.x >> 5) + (threadIdx.x >> 5);
    const int f = threadIdx.x & 31;
    if (e >= E + N) return;
    int s, d;
    if (e < E) { s = src[e]; d = dst[e]; }
    else       { s = e - E; d = s; }
    const float norm = dis[s] * dis[d];
    const float v = xw[(size_t)s * 32 + f] * norm;
    atomicAdd(&out[(size_t)d * 32 + f], v);
}

__global__ void bias_tanh32(const float* __restrict__ in,
                            const float* __restrict__ b,
                            float* __restrict__ out, int total) {
    int i = blockIdx.x * blockDim.x + threadIdx.x;
    if (i < total) out[i] = tanhf(in[i] + b[i & 31]);
}

// ---------------------------------------------------------------------------
// Host-side orchestration
// ---------------------------------------------------------------------------
extern "C" void kernel_launch(void* const* d_in, const int* in_sizes, int n_in,
                              void* d_out, int out_size, void* d_ws, size_t ws_size,
                              hipStream_t stream) {
    const float* x  = (const float*)d_in[0];
    const int*   ei = (const int*)  d_in[1];
    const float* W1 = (const float*)d_in[2];
    const float* b1 = (const float*)d_in[3];
    const float* W2 = (const float*)d_in[4];
    const float* b2 = (const float*)d_in[5];
    const float* Wc = (const float*)d_in[6];
    const float* bc = (const float*)d_in[7];
    float* out = (float*)d_out;

    const int H = in_sizes[3];          // 32
    const int C = in_sizes[7];          // 16
    const int F = in_sizes[2] / H;      // 128
    const int N = in_sizes[0] / F;      // 50000
    const int E = in_sizes[1] / 2;      // 1600000

    const int* src = ei;
    const int* dst = ei + E;

    // Workspace: dis[N] | buf1[N*H] | buf2[N*H]   (~13 MB)
    float* dis  = (float*)d_ws;
    float* buf1 = dis + (((size_t)N + 63) & ~(size_t)63);
    float* buf2 = buf1 + (size_t)N * H;
    float* hout = out + (size_t)N * C;  // second tuple output: h [N x H]

    const int T   = 256;            // 8 waves / block
    const int WPB = T / 32;
    const int NH  = N * H;
    const int tiles = (N + 15) / 16;
    const int etot  = E + N;

    // deg_inv_sqrt: deg = 1 (self-loop) + sum over edges on dst, then rsqrt
    fill_f32 <<<(N + T - 1) / T, T, 0, stream>>>(dis, 1.0f, N);
    deg_accum<<<(E + T - 1) / T, T, 0, stream>>>(dst, dis, E);
    deg_rsqrt<<<(N + T - 1) / T, T, 0, stream>>>(dis, N);

    // ---- Layer 1: xw = x@W1 ; agg ; tanh(+b1) -> buf1 ----
    gemm_wmma_f32<128, 2><<<(tiles + WPB - 1) / WPB, T, 0, stream>>>(x, W1, nullptr, buf1, N);
    fill_f32   <<<(NH + T - 1) / T, T, 0, stream>>>(buf2, 0.0f, NH);
    aggregate32<<<(etot + WPB - 1) / WPB, T, 0, stream>>>(buf1, src, dst, dis, buf2, E, N);
    bias_tanh32<<<(NH + T - 1) / T, T, 0, stream>>>(buf2, b1, buf1, NH);

    // ---- Layer 2: xw2 = h1@W2 ; agg ; tanh(+b2) -> hout (d_out slice) ----
    gemm_wmma_f32<32, 2><<<(tiles + WPB - 1) / WPB, T, 0, stream>>>(buf1, W2, nullptr, buf2, N);
    fill_f32   <<<(NH + T - 1) / T, T, 0, stream>>>(buf1, 0.0f, NH);
    aggregate32<<<(etot + WPB - 1) / WPB, T, 0, stream>>>(buf2, src, dst, dis, buf1, E, N);
    bias_tanh32<<<(NH + T - 1) / T, T, 0, stream>>>(buf1, b2, hout, NH);

    // ---- Classifier: out = h2@Wc + bc ----
    gemm_wmma_f32<32, 1><<<(tiles + WPB - 1) / WPB, T, 0, stream>>>(hout, Wc, bc, out, N);
}